// Elastic_67577015435856
// MI455X (gfx1250) — compile-verified
//
#include <hip/hip_runtime.h>
#include <stdint.h>

// ---------------------------------------------------------------------------
// Problem constants (match the JAX reference)
// ---------------------------------------------------------------------------
namespace {
constexpr int NY = 512, NX = 512, B = 2, NT = 32, NREC = 128;
constexpr float DT  = 0.001f;
constexpr float RDY = 1.0f / 4.0f;
constexpr float RDX = 1.0f / 4.0f;
constexpr float C1f = 9.0f / 8.0f;
constexpr float C2f = -1.0f / 24.0f;

constexpr int TDIM     = 32;          // 32x32 output tile per block
constexpr int LH       = 36;          // staged rows: y0-2 .. y0+33
constexpr int LW       = 40;          // staged cols: x0-4 .. x0+35 (16B chunks)
constexpr int YOFF     = 2;           // y halo offset in LDS
constexpr int XOFF     = 4;           // x halo offset in LDS (aligned)
constexpr int CHUNKS   = LW / 4;      // 10 x b128 chunks per row
constexpr int NCHUNK   = LH * CHUNKS; // 360 chunks per field
constexpr int NTHREADS = 256;         // 8 wave32 per block
}

// ---------------------------------------------------------------------------
// CDNA5 async global->LDS staging (b128 granularity). Falls back to plain
// vector copies if the builtins are unavailable.
// ---------------------------------------------------------------------------
#if defined(__HIP_DEVICE_COMPILE__) && \
    __has_builtin(__builtin_amdgcn_global_load_async_to_lds_b128)
#define USE_ASYNC_LDS 1
#else
#define USE_ASYNC_LDS 0
#endif

#if USE_ASYNC_LDS
typedef int v4i __attribute__((ext_vector_type(4)));
typedef __attribute__((address_space(1))) v4i gas_v4i;  // global (AS1) int4
typedef __attribute__((address_space(3))) v4i las_v4i;  // LDS (AS3) int4
#endif

// Stage an LH x LW tile (zero-filled halo) of field g into LDS.
// gy0 = y0 - YOFF, gx0 = x0 - XOFF (gx0 is a multiple of 4 => every 16B chunk
// is either fully inside [0,NX) or fully outside, since NX % 4 == 0).
__device__ __forceinline__ void stage_tile(float* __restrict__ lds,
                                           const float* __restrict__ g,
                                           int gy0, int gx0, int tid)
{
  for (int i = tid; i < NCHUNK; i += NTHREADS) {
    const int r  = i / CHUNKS;
    const int c  = i - r * CHUNKS;
    const int gy = gy0 + r;
    const int gx = gx0 + 4 * c;
    float* dst = lds + r * LW + 4 * c;
    if (gy >= 0 && gy < NY && gx >= 0 && gx < NX) {
#if USE_ASYNC_LDS
      float* gp = const_cast<float*>(g + (size_t)gy * NX + gx);
      __builtin_amdgcn_global_load_async_to_lds_b128(
          (gas_v4i*)gp, (las_v4i*)dst, /*offset=*/0, /*cpol=*/0);
#else
      *reinterpret_cast<float4*>(dst) =
          *reinterpret_cast<const float4*>(g + (size_t)gy * NX + gx);
#endif
    } else {
      *reinterpret_cast<float4*>(dst) = make_float4(0.f, 0.f, 0.f, 0.f);
    }
  }
}

__device__ __forceinline__ void wait_stage()
{
#if USE_ASYNC_LDS
  asm volatile("s_wait_asynccnt 0" ::: "memory");  // drain ASYNCcnt before use
#endif
  __syncthreads();
}

// ---------------------------------------------------------------------------
// Velocity update: reads syy/sxy/sxx (staged in LDS), updates PML stress
// memories and vy/vx.
// ---------------------------------------------------------------------------
__global__ void __launch_bounds__(NTHREADS)
vel_kernel(float* __restrict__ vy, float* __restrict__ vx,
           const float* __restrict__ syy, const float* __restrict__ sxy,
           const float* __restrict__ sxx,
           float* __restrict__ m_syyy, float* __restrict__ m_sxyy,
           float* __restrict__ m_sxyx, float* __restrict__ m_sxxx,
           const float* __restrict__ buoy_y, const float* __restrict__ buoy_x,
           const float* __restrict__ ay, const float* __restrict__ ayh,
           const float* __restrict__ ax, const float* __restrict__ axh,
           const float* __restrict__ by, const float* __restrict__ byh,
           const float* __restrict__ bx, const float* __restrict__ bxh)
{
  __shared__ __align__(16) float s_syy[LH * LW];
  __shared__ __align__(16) float s_sxy[LH * LW];
  __shared__ __align__(16) float s_sxx[LH * LW];

  const int b   = blockIdx.z;
  const int y0  = blockIdx.y * TDIM;
  const int x0  = blockIdx.x * TDIM;
  const int tid = threadIdx.x;
  const size_t base = (size_t)b * NY * NX;

  stage_tile(s_syy, syy + base, y0 - YOFF, x0 - XOFF, tid);
  stage_tile(s_sxy, sxy + base, y0 - YOFF, x0 - XOFF, tid);
  stage_tile(s_sxx, sxx + base, y0 - YOFF, x0 - XOFF, tid);
  wait_stage();

  const int lx  = tid & (TDIM - 1);
  const int lyw = tid >> 5;  // 0..7

#pragma unroll
  for (int k = 0; k < 4; ++k) {
    const int ly = lyw + 8 * k;
    const int y  = y0 + ly;
    const int x  = x0 + lx;
    const int c  = (ly + YOFF) * LW + (lx + XOFF);

    // dsyydy = diffh1_y(syy):  C1*(f[y]-f[y-1]) + C2*(f[y+1]-f[y-2])
    float dsyydy = (C1f * (s_syy[c] - s_syy[c - LW]) +
                    C2f * (s_syy[c + LW] - s_syy[c - 2 * LW])) * RDY;
    // dsxydx = diff1_x(sxy):   C1*(f[x+1]-f[x]) + C2*(f[x+2]-f[x-1])
    float dsxydx = (C1f * (s_sxy[c + 1] - s_sxy[c]) +
                    C2f * (s_sxy[c + 2] - s_sxy[c - 1])) * RDX;
    // dsxydy = diff1_y(sxy)
    float dsxydy = (C1f * (s_sxy[c + LW] - s_sxy[c]) +
                    C2f * (s_sxy[c + 2 * LW] - s_sxy[c - LW])) * RDY;
    // dsxxdx = diffh1_x(sxx)
    float dsxxdx = (C1f * (s_sxx[c] - s_sxx[c - 1]) +
                    C2f * (s_sxx[c + 1] - s_sxx[c - 2])) * RDX;

    const size_t g = base + (size_t)y * NX + x;
    const int    p = y * NX + x;

    float m0 = ayh[y] * m_syyy[g] + byh[y] * dsyydy; m_syyy[g] = m0; dsyydy += m0;
    float m1 = ax[x]  * m_sxyx[g] + bx[x]  * dsxydx; m_sxyx[g] = m1; dsxydx += m1;
    float m2 = ay[y]  * m_sxyy[g] + by[y]  * dsxydy; m_sxyy[g] = m2; dsxydy += m2;
    float m3 = axh[x] * m_sxxx[g] + bxh[x] * dsxxdx; m_sxxx[g] = m3; dsxxdx += m3;

    vy[g] += buoy_y[p] * DT * (dsyydy + dsxydx);
    vx[g] += buoy_x[p] * DT * (dsxydy + dsxxdx);
  }
}

// ---------------------------------------------------------------------------
// Stress update: reads vy/vx (staged in LDS), updates PML velocity memories
// and syy/sxx/sxy.
// ---------------------------------------------------------------------------
__global__ void __launch_bounds__(NTHREADS)
stress_kernel(const float* __restrict__ vy, const float* __restrict__ vx,
              float* __restrict__ syy, float* __restrict__ sxy,
              float* __restrict__ sxx,
              float* __restrict__ m_vyy, float* __restrict__ m_vyx,
              float* __restrict__ m_vxy, float* __restrict__ m_vxx,
              const float* __restrict__ lamb, const float* __restrict__ mu,
              const float* __restrict__ mu_yx,
              const float* __restrict__ ay, const float* __restrict__ ayh,
              const float* __restrict__ ax, const float* __restrict__ axh,
              const float* __restrict__ by, const float* __restrict__ byh,
              const float* __restrict__ bx, const float* __restrict__ bxh)
{
  __shared__ __align__(16) float s_vy[LH * LW];
  __shared__ __align__(16) float s_vx[LH * LW];

  const int b   = blockIdx.z;
  const int y0  = blockIdx.y * TDIM;
  const int x0  = blockIdx.x * TDIM;
  const int tid = threadIdx.x;
  const size_t base = (size_t)b * NY * NX;

  stage_tile(s_vy, vy + base, y0 - YOFF, x0 - XOFF, tid);
  stage_tile(s_vx, vx + base, y0 - YOFF, x0 - XOFF, tid);
  wait_stage();

  const int lx  = tid & (TDIM - 1);
  const int lyw = tid >> 5;

#pragma unroll
  for (int k = 0; k < 4; ++k) {
    const int ly = lyw + 8 * k;
    const int y  = y0 + ly;
    const int x  = x0 + lx;
    const int c  = (ly + YOFF) * LW + (lx + XOFF);

    // dvydy = diff1_y(vy), dvxdx = diff1_x(vx)
    float dvydy = (C1f * (s_vy[c + LW] - s_vy[c]) +
                   C2f * (s_vy[c + 2 * LW] - s_vy[c - LW])) * RDY;
    float dvxdx = (C1f * (s_vx[c + 1] - s_vx[c]) +
                   C2f * (s_vx[c + 2] - s_vx[c - 1])) * RDX;

    const size_t g = base + (size_t)y * NX + x;
    const int    p = y * NX + x;

    float m0 = ay[y] * m_vyy[g] + by[y] * dvydy; m_vyy[g] = m0; dvydy += m0;
    float m1 = ax[x] * m_vxx[g] + bx[x] * dvxdx; m_vxx[g] = m1; dvxdx += m1;

    const float L    = lamb[p];
    const float lp2m = L + 2.0f * mu[p];
    syy[g] += DT * (lp2m * dvydy + L * dvxdx);
    sxx[g] += DT * (lp2m * dvxdx + L * dvydy);

    // dvydx = diffh1_x(vy), dvxdy = diffh1_y(vx)
    float dvydx = (C1f * (s_vy[c] - s_vy[c - 1]) +
                   C2f * (s_vy[c + 1] - s_vy[c - 2])) * RDX;
    float dvxdy = (C1f * (s_vx[c] - s_vx[c - LW]) +
                   C2f * (s_vx[c + LW] - s_vx[c - 2 * LW])) * RDY;

    float m2 = ayh[y] * m_vxy[g] + byh[y] * dvxdy; m_vxy[g] = m2; dvxdy += m2;
    float m3 = axh[x] * m_vyx[g] + bxh[x] * dvydx; m_vyx[g] = m3; dvydx += m3;

    sxy[g] += DT * mu_yx[p] * (dvydx + dvxdy);
  }
}

// ---------------------------------------------------------------------------
// Source injection (into vy/vx) then receiver sampling of vy. One block so the
// inject -> record ordering of the reference is preserved with __syncthreads.
// ---------------------------------------------------------------------------
__global__ void __launch_bounds__(NTHREADS)
src_rec_kernel(float* __restrict__ vy, float* __restrict__ vx,
               const float* __restrict__ amp_y, const float* __restrict__ amp_x,
               const int* __restrict__ src_y_i, const int* __restrict__ src_x_i,
               const int* __restrict__ rec_y_i, float* __restrict__ rec, int t)
{
  const int tid = threadIdx.x;
  if (tid < B) {
    vy[(size_t)tid * NY * NX + src_y_i[tid]] += amp_y[t * B + tid];
    vx[(size_t)tid * NY * NX + src_x_i[tid]] += amp_x[t * B + tid];
  }
  __syncthreads();
  const int bsh = tid / NREC;  // blockDim.x == B*NREC == 256
  if (bsh < B) {
    rec[(size_t)t * B * NREC + tid] =
        vy[(size_t)bsh * NY * NX + rec_y_i[tid]];
  }
}

// recs[1:] and recs[:-1] averaged -> output (NT-1, B, NREC)
__global__ void avg_kernel(const float* __restrict__ rec, float* __restrict__ out,
                           int n)
{
  const int i = blockIdx.x * blockDim.x + threadIdx.x;
  if (i < n) out[i] = 0.5f * (rec[i] + rec[i + B * NREC]);
}

// ---------------------------------------------------------------------------
// Host-side launcher (graph-capture safe: only kernel launches + MemsetAsync).
// ---------------------------------------------------------------------------
extern "C" void kernel_launch(void* const* d_in, const int* in_sizes, int n_in,
                              void* d_out, int out_size, void* d_ws, size_t ws_size,
                              hipStream_t stream)
{
  (void)in_sizes; (void)n_in; (void)ws_size;

  const float* lamb    = (const float*)d_in[0];
  const float* mu      = (const float*)d_in[1];
  const float* mu_yx   = (const float*)d_in[2];
  const float* buoy_y  = (const float*)d_in[3];
  const float* buoy_x  = (const float*)d_in[4];
  const float* amp_y   = (const float*)d_in[5];   // (NT, B, 1)
  const float* amp_x   = (const float*)d_in[6];
  const float* ay      = (const float*)d_in[7];   // (NY)
  const float* ayh     = (const float*)d_in[8];
  const float* ax      = (const float*)d_in[9];   // (NX)
  const float* axh     = (const float*)d_in[10];
  const float* by      = (const float*)d_in[11];
  const float* byh     = (const float*)d_in[12];
  const float* bx      = (const float*)d_in[13];
  const float* bxh     = (const float*)d_in[14];
  const int*   src_y_i = (const int*)d_in[15];    // (B,1)
  const int*   src_x_i = (const int*)d_in[16];
  const int*   rec_y_i = (const int*)d_in[17];    // (B,NREC)

  const size_t F = (size_t)B * NY * NX;  // elements per state field
  float* W = (float*)d_ws;
  float* vy     = W + 0 * F;
  float* vx     = W + 1 * F;
  float* syy    = W + 2 * F;
  float* sxy    = W + 3 * F;
  float* sxx    = W + 4 * F;
  float* m_vyy  = W + 5 * F;
  float* m_vyx  = W + 6 * F;
  float* m_vxy  = W + 7 * F;
  float* m_vxx  = W + 8 * F;
  float* m_syyy = W + 9 * F;
  float* m_sxyy = W + 10 * F;
  float* m_sxyx = W + 11 * F;
  float* m_sxxx = W + 12 * F;
  float* rec    = W + 13 * F;            // (NT, B, NREC)

  const size_t zero_bytes = (13 * F + (size_t)NT * B * NREC) * sizeof(float);
  (void)hipMemsetAsync(d_ws, 0, zero_bytes, stream);

  dim3 grid(NX / TDIM, NY / TDIM, B);    // 16 x 16 x 2 blocks
  dim3 blk(NTHREADS);

  for (int t = 0; t < NT; ++t) {
    vel_kernel<<<grid, blk, 0, stream>>>(
        vy, vx, syy, sxy, sxx, m_syyy, m_sxyy, m_sxyx, m_sxxx,
        buoy_y, buoy_x, ay, ayh, ax, axh, by, byh, bx, bxh);
    src_rec_kernel<<<1, blk, 0, stream>>>(
        vy, vx, amp_y, amp_x, src_y_i, src_x_i, rec_y_i, rec, t);
    stress_kernel<<<grid, blk, 0, stream>>>(
        vy, vx, syy, sxy, sxx, m_vyy, m_vyx, m_vxy, m_vxx,
        lamb, mu, mu_yx, ay, ayh, ax, axh, by, byh, bx, bxh);
  }

  int n = (NT - 1) * B * NREC;
  if (n > out_size) n = out_size;
  avg_kernel<<<(n + NTHREADS - 1) / NTHREADS, blk, 0, stream>>>(
      rec, (float*)d_out, n);
}